// PatchFusionAttention_1872605741733
// MI455X (gfx1250) — compile-verified
//
#include <hip/hip_runtime.h>

// ---------------------------------------------------------------------------
// PatchFusionAttention for MI455X (gfx1250, wave32).
// B=32, T=H*W=1024 tokens, F=768 features. Only one attention row per batch
// is consumed, and the softmax denominator cancels against the final
// renormalization, so the O(T^2 F) attention collapses to two GEMVs per
// batch. Memory-bound: ~3 passes over x (96 MB); L2 (192 MB) holds x, so
// HBM traffic ~96 MB (~4-5 us at 23.3 TB/s). Score GEMV runs on
// v_wmma_f32_16x16x4_f32 (16 tokens per wave per tile, query in column 0).
// ---------------------------------------------------------------------------

#define BATCH 32
#define TOK   1024
#define FEAT  768

typedef __attribute__((ext_vector_type(2))) float v2f;
typedef __attribute__((ext_vector_type(8))) float v8f;

// ---------------------------------------------------------------------------
// K1: per-token classifier prob + inverse L2 norm. One wave per token,
// 8 tokens per 256-thread block. float4 streaming loads (b128).
// ---------------------------------------------------------------------------
__global__ __launch_bounds__(256) void k_probs_norms(
    const float* __restrict__ x, const float* __restrict__ fc_w,
    const float* __restrict__ fc_b, float* __restrict__ probs,
    float* __restrict__ invn)
{
    const int wave = threadIdx.x >> 5;
    const int lane = threadIdx.x & 31;
    const int tok  = blockIdx.x * 8 + wave;          // [0, BATCH*TOK)

    const float4* xr = (const float4*)(x + (size_t)tok * FEAT);
    const float4* w0 = (const float4*)(fc_w);
    const float4* w1 = (const float4*)(fc_w + FEAT);

    float s0 = 0.f, s1 = 0.f, ss = 0.f;
#pragma unroll
    for (int i = 0; i < 6; ++i) {                    // 6*32 = 192 float4 = 768
        const int j = lane + 32 * i;
        const float4 v = xr[j];
        const float4 a = w0[j];
        const float4 b = w1[j];
        s0 += v.x * a.x + v.y * a.y + v.z * a.z + v.w * a.w;
        s1 += v.x * b.x + v.y * b.y + v.z * b.z + v.w * b.w;
        ss += v.x * v.x + v.y * v.y + v.z * v.z + v.w * v.w;
    }
#pragma unroll
    for (int off = 16; off; off >>= 1) {
        s0 += __shfl_xor(s0, off, 32);
        s1 += __shfl_xor(s1, off, 32);
        ss += __shfl_xor(ss, off, 32);
    }
    if (lane == 0) {
        const float l0 = s0 + fc_b[0];
        const float l1 = s1 + fc_b[1];
        probs[tok] = 1.0f / (1.0f + expf(l0 - l1));  // softmax(...)[1]
        invn[tok]  = rsqrtf(ss);
    }
}

// ---------------------------------------------------------------------------
// K2: per-batch argmax over 1024 probs, first-occurrence tie-break (jnp).
// ---------------------------------------------------------------------------
__global__ __launch_bounds__(256) void k_argmax(
    const float* __restrict__ probs, int* __restrict__ idx)
{
    __shared__ float sv[256];
    __shared__ int   si[256];
    const int b = blockIdx.x, tid = threadIdx.x;
    const float* p = probs + b * TOK;

    float bv = -1e30f; int bi = 0;
    for (int t = tid; t < TOK; t += 256) {           // ascending: strict '>'
        const float v = p[t];                        // keeps smallest index
        if (v > bv) { bv = v; bi = t; }
    }
    sv[tid] = bv; si[tid] = bi;
    __syncthreads();
    for (int s = 128; s; s >>= 1) {
        if (tid < s) {
            const float ov = sv[tid + s]; const int oi = si[tid + s];
            if (ov > sv[tid] || (ov == sv[tid] && oi < si[tid])) {
                sv[tid] = ov; si[tid] = oi;
            }
        }
        __syncthreads();
    }
    if (tid == 0) idx[b] = si[0];
}

// ---------------------------------------------------------------------------
// K3: one block per batch (1024 threads = 32 waves).
//   a) query row -> LDS
//   b) scores[t] = q . x_t via v_wmma_f32_16x16x4_f32 (wave owns 32 tokens)
//   c) cosine scale, block max, e = exp(s-m)*p, block sum (softmax denom
//      cancels against final renorm: w = e*p / sum(e*p))
//   d) x_fused = sum_t w_t x_t  (coalesced, one feature per thread)
//   e) 2-class head
// ---------------------------------------------------------------------------
__global__ __launch_bounds__(1024) void k_fuse(
    const float* __restrict__ x, const float* __restrict__ probs,
    const float* __restrict__ invn, const int* __restrict__ idxbuf,
    const float* __restrict__ fcf_w, const float* __restrict__ fcf_b,
    float* __restrict__ out)
{
    __shared__ __align__(16) float q[FEAT];    // query token
    __shared__ __align__(16) float sc[TOK];    // dot -> score -> e*p
    __shared__ __align__(16) float xf[FEAT];   // fused feature vector
    __shared__ float red[32];
    __shared__ float s_bcast[2];               // [0]=max, [1]=1/sum
    __shared__ int   s_idx;

    const int b    = blockIdx.x;
    const int tid  = threadIdx.x;
    const int wave = tid >> 5;
    const int lane = tid & 31;
    const float* xb = x + (size_t)b * TOK * FEAT;

    if (tid == 0) s_idx = idxbuf[b];
    __syncthreads();
    const int qi = s_idx;
    for (int f = tid; f < FEAT; f += 1024) q[f] = xb[(size_t)qi * FEAT + f];
    const float q_invn = invn[b * TOK + qi];
    __syncthreads();

    // ---- b) WMMA score GEMV: wave handles tokens [wave*32, wave*32+32) ----
    {
        const int col  = (lane >> 4) * 2;      // K sub-pair: 0 or 2
        const int rowl = lane & 15;            // A-matrix row within tile
        const bool keep = (rowl == 0);         // only column 0 of B is live
        for (int g = 0; g < 2; ++g) {
            const int t0 = wave * 32 + g * 16;
            const float* arow = xb + (size_t)(t0 + rowl) * FEAT + col;
            v8f c = {0.f, 0.f, 0.f, 0.f, 0.f, 0.f, 0.f, 0.f};
#pragma unroll 4
            for (int k0 = 0; k0 < FEAT; k0 += 4) {
                // A: 16x4 fp32 tile; lane l holds row l&15, K = 2*(l>>4)+{0,1}
                const v2f a = *reinterpret_cast<const v2f*>(arow + k0);
                // B: 4x16 with only column 0 nonzero (query chunk)
                const v2f qv = *reinterpret_cast<const v2f*>(&q[k0 + col]);
                v2f bm;
                bm.x = keep ? qv.x : 0.0f;
                bm.y = keep ? qv.y : 0.0f;
                c = __builtin_amdgcn_wmma_f32_16x16x4_f32(
                        false, a, false, bm, (short)0, c, false, false);
            }
            // Column 0 of C: lane0 -> rows 0..7, lane16 -> rows 8..15
            if (keep) {
                const int base = t0 + (lane >> 4) * 8;
#pragma unroll
                for (int j = 0; j < 8; ++j) sc[base + j] = c[j];
            }
        }
    }
    __syncthreads();

    // ---- c) cosine scale + stable softmax-numerator * prob ----
    const float* inb = invn + b * TOK;
    const float* pb  = probs + b * TOK;

    float s = sc[tid] * inb[tid] * q_invn;     // one token per thread
    sc[tid] = s;
    float m = s;
#pragma unroll
    for (int off = 16; off; off >>= 1) m = fmaxf(m, __shfl_xor(m, off, 32));
    if (lane == 0) red[wave] = m;
    __syncthreads();
    if (wave == 0) {
        float r = red[lane];
#pragma unroll
        for (int off = 16; off; off >>= 1) r = fmaxf(r, __shfl_xor(r, off, 32));
        if (lane == 0) s_bcast[0] = r;
    }
    __syncthreads();
    m = s_bcast[0];
    __syncthreads();

    float e = expf(sc[tid] - m) * pb[tid];
    sc[tid] = e;                               // unnormalized weight
    float ssum = e;
#pragma unroll
    for (int off = 16; off; off >>= 1) ssum += __shfl_xor(ssum, off, 32);
    if (lane == 0) red[wave] = ssum;
    __syncthreads();
    if (wave == 0) {
        float r = red[lane];
#pragma unroll
        for (int off = 16; off; off >>= 1) r += __shfl_xor(r, off, 32);
        if (lane == 0) s_bcast[1] = 1.0f / r;
    }
    __syncthreads();
    const float rS = s_bcast[1];
    __syncthreads();

    // ---- d) x_fused[f] = rS * sum_t w_t * x[t][f]  (coalesced over f) ----
    if (tid < FEAT) {
        float acc = 0.f;
        for (int t = 0; t < TOK; ++t) {
            const float wl = sc[t];            // LDS broadcast
            const float* row = xb + (size_t)t * FEAT;
            acc += wl * row[tid];
            const int tn = (t + 1 < TOK) ? t + 1 : t;
            __builtin_prefetch(xb + (size_t)tn * FEAT + tid, 0, 0);
        }
        xf[tid] = acc * rS;
    }
    __syncthreads();

    // ---- e) 2-class head: wave 0 -> class 0, wave 1 -> class 1 ----
    if (wave < 2) {
        const float* wrow = fcf_w + wave * FEAT;
        float acc = 0.f;
#pragma unroll
        for (int i = 0; i < FEAT / 32; ++i) {
            const int f = lane + 32 * i;
            acc += xf[f] * wrow[f];
        }
#pragma unroll
        for (int off = 16; off; off >>= 1) acc += __shfl_xor(acc, off, 32);
        if (lane == 0) out[b * 2 + wave] = acc + fcf_b[wave];
    }
}

// ---------------------------------------------------------------------------
extern "C" void kernel_launch(void* const* d_in, const int* in_sizes, int n_in,
                              void* d_out, int out_size, void* d_ws, size_t ws_size,
                              hipStream_t stream) {
    const float* x    = (const float*)d_in[0];   // [32,32,32,768]
    const float* fcw  = (const float*)d_in[1];   // [2,768]
    const float* fcb  = (const float*)d_in[2];   // [2]
    const float* ffw  = (const float*)d_in[3];   // [2,768]
    const float* ffb  = (const float*)d_in[4];   // [2]
    float* out = (float*)d_out;                  // [32,2] fp32

    char* ws = (char*)d_ws;
    float* probs = (float*)(ws);                 // 32768 floats (128 KB)
    float* invn  = (float*)(ws + 131072);        // 32768 floats (128 KB)
    int*   idx   = (int*)(ws + 262144);          // 32 ints

    k_probs_norms<<<dim3(BATCH * TOK / 8), dim3(256), 0, stream>>>(
        x, fcw, fcb, probs, invn);
    k_argmax<<<dim3(BATCH), dim3(256), 0, stream>>>(probs, idx);
    k_fuse<<<dim3(BATCH), dim3(1024), 0, stream>>>(
        x, probs, invn, idx, ffw, ffb, out);
}